// L0RegularizedLinear_87058987090490
// MI455X (gfx1250) — compile-verified
//
#include <hip/hip_runtime.h>
#include <stdint.h>

#define B_    32
#define OUT_  2048
#define IN_   2048
#define WAVES 4
#define CK    32                    // i-columns staged per chunk
#define PITCH 36                    // padded LDS row pitch (floats) -> bank-conflict-free
#define ROWS  16
#define TILE_F (ROWS * PITCH)       // 576 floats per array per buffer
#define BUF_F  (3 * TILE_F)         // u,w,la per buffer = 1728 floats
#define WAVE_F (2 * BUF_F)          // double buffered = 3456 floats / wave
#define XS_OFF (WAVES * WAVE_F)     // 13824: x row (2048 floats)
#define RS_OFF (XS_OFF + IN_)       // 15872: per-wave partials (4*16)
#define SMEM_F (RS_OFF + WAVES * 16)// 15936 floats = 63744 B

typedef __attribute__((ext_vector_type(2))) float v2f;
typedef __attribute__((ext_vector_type(4))) float v4f;
typedef __attribute__((ext_vector_type(8))) float v8f;

// LDS byte offset of a __shared__ address (generic -> addrspace(3) -> u32)
__device__ __forceinline__ uint32_t lds_addr32(const void* p) {
  return (uint32_t)(uintptr_t)(const __attribute__((address_space(3))) char*)(const char*)p;
}

// CDNA5 async copy: global -> LDS, 16B per lane, tracked by ASYNCcnt
__device__ __forceinline__ void async_copy_b128(uint32_t lds_byte, uint32_t goff_byte,
                                                uint64_t sbase) {
  asm volatile("global_load_async_to_lds_b128 %0, %1, %2"
               :: "v"(lds_byte), "v"(goff_byte), "s"(sbase)
               : "memory");
}
#define WAIT_ASYNC(n) asm volatile("s_wait_asynccnt " #n ::: "memory")

// hard-concrete gate: z = clip(sigmoid((ln u - ln(1-u) + la)/ (2/3)) * 1.2 - 0.1, 0, 1)
//                       = clip(0.5 + 0.6 * tanh(0.75*(ln u - ln(1-u) + la)), 0, 1)
__device__ __forceinline__ float gate_z(float uu, float la) {
  // v_log_f32 is log2; fold ln2 into constants
  float dl = __builtin_amdgcn_logf(uu) - __builtin_amdgcn_logf(1.0f - uu); // log2(u/(1-u))
#if __has_builtin(__builtin_amdgcn_tanhf)
  // 0.75*ln2 = 0.519860385419958
  float th = __builtin_amdgcn_tanhf(__builtin_fmaf(dl, 0.519860385419958f, 0.75f * la));
  float z  = __builtin_fmaf(th, 0.6f, 0.5f);
#else
  // exp(-t) = exp2(-1.5*dl - (1.5/ln2)*la), 1.5/ln2 = 2.1640425613334453
  float e = __builtin_amdgcn_exp2f(__builtin_fmaf(la, -2.1640425613334453f, -1.5f * dl));
  float s = __builtin_amdgcn_rcpf(1.0f + e);
  float z = __builtin_fmaf(s, 1.2f, -0.1f);
#endif
  return fminf(fmaxf(z, 0.0f), 1.0f);
}

extern "C" __global__ void __launch_bounds__(128)
l0_masked_linear_kernel(const float* __restrict__ x,
                        const float* __restrict__ u,
                        const float* __restrict__ w,
                        const float* __restrict__ la,
                        const float* __restrict__ bias,
                        float* __restrict__ out) {
  __shared__ __align__(16) float smem[SMEM_F];

  const uint32_t tid   = threadIdx.x;
  const uint32_t lane  = tid & 31u;
  const uint32_t wv    = tid >> 5;
  const uint32_t b     = blockIdx.x >> 7;   // OUT_/16 == 128 tiles
  const uint32_t ot    = blockIdx.x & 127u;
  const uint32_t orow0 = ot * 16u;
  const uint32_t iw0   = wv * (IN_ / WAVES);  // 512 i's per wave

  // ---- stage x[b,:] into LDS (coalesced b128) ----
  {
    const float* xb = x + (size_t)b * IN_;
#pragma unroll
    for (int p = 0; p < IN_ / (128 * 4); ++p) {
      int i = (p * 128 + (int)tid) * 4;
      *(v4f*)&smem[XS_OFF + i] = *(const v4f*)&xb[i];
    }
  }

  // ---- per-lane async-copy descriptors: 4 rows/instr, 8 lanes/row ----
  const uint32_t r4 = lane >> 3;   // 0..3 (row within group of 4)
  const uint32_t c8 = lane & 7u;   // 0..7 (16B column slot)
  uint32_t gU[4], gW[4], lR[4];
  const uint32_t smem0 = lds_addr32(smem);
#pragma unroll
  for (int g = 0; g < 4; ++g) {
    uint32_t row = (uint32_t)g * 4u + r4;
    gU[g] = (((b * OUT_ + orow0 + row) * (uint32_t)IN_) + iw0 + c8 * 4u) * 4u;
    gW[g] = ((((orow0 + row)) * (uint32_t)IN_) + iw0 + c8 * 4u) * 4u;
    lR[g] = smem0 + (wv * WAVE_F + row * PITCH + c8 * 4u) * 4u;
  }
  const uint64_t ub64 = (uint64_t)(uintptr_t)u;
  const uint64_t wb64 = (uint64_t)(uintptr_t)w;
  const uint64_t lb64 = (uint64_t)(uintptr_t)la;

  const int NCH = (IN_ / WAVES) / CK;  // 16 chunks, double-buffered

  // prologue: issue chunk 0 (12 async instrs: u, w, la)
#pragma unroll
  for (int g = 0; g < 4; ++g) {
    async_copy_b128(lR[g],                  gU[g], ub64);
    async_copy_b128(lR[g] + TILE_F * 4,     gW[g], wb64);
    async_copy_b128(lR[g] + 2 * TILE_F * 4, gW[g], lb64);
  }

  __syncthreads();  // x visible to all waves

  // ---- WMMA setup: A = masked products, B = ones in column 0 ----
  const uint32_t row  = lane & 15u;         // A-matrix M = lane&15
  const uint32_t coff = (lane >> 4) * 2u;   // K-pair: lanes<16 -> K0,1 ; lanes>=16 -> K2,3
  const float bones = (lane == 0u || lane == 16u) ? 1.0f : 0.0f; // B[k,0]=1
  v2f bv; bv.x = bones; bv.y = bones;
  v8f acc = {0.f, 0.f, 0.f, 0.f, 0.f, 0.f, 0.f, 0.f};

  for (int c = 0; c < NCH; ++c) {
    if (c + 1 < NCH) {  // prefetch next chunk into the other buffer
      uint32_t gadd = (uint32_t)(c + 1) * (CK * 4u);
      uint32_t ladd = (uint32_t)((c + 1) & 1) * (BUF_F * 4u);
#pragma unroll
      for (int g = 0; g < 4; ++g) {
        async_copy_b128(lR[g] + ladd,                  gU[g] + gadd, ub64);
        async_copy_b128(lR[g] + ladd + TILE_F * 4,     gW[g] + gadd, wb64);
        async_copy_b128(lR[g] + ladd + 2 * TILE_F * 4, gW[g] + gadd, lb64);
      }
      WAIT_ASYNC(12);   // chunk c complete; chunk c+1 (12 ops) in flight
    } else {
      WAIT_ASYNC(0);
    }

    const float* ub = &smem[wv * WAVE_F + (uint32_t)(c & 1) * BUF_F];
    const float* wb = ub + TILE_F;
    const float* lb = ub + 2 * TILE_F;
    const float* xs = &smem[XS_OFF + iw0 + (uint32_t)c * CK];

#pragma unroll
    for (int k = 0; k < CK / 4; ++k) {
      uint32_t col = (uint32_t)k * 4u + coff;     // this lane's K-pair
      v2f uu = *(const v2f*)&ub[row * PITCH + col];  // conflict-free ds_load_b64
      v2f ww = *(const v2f*)&wb[row * PITCH + col];
      v2f ll = *(const v2f*)&lb[row * PITCH + col];
      v2f xx = *(const v2f*)&xs[col];                // broadcast read
      v2f a;
      a.x = ww.x * xx.x * gate_z(uu.x, ll.x);
      a.y = ww.y * xx.y * gate_z(uu.y, ll.y);
      // D[:,0] += sum_k A[:,k] : full-f32 reduction on the matrix core
      acc = __builtin_amdgcn_wmma_f32_16x16x4_f32(
          false, a, false, bv, (short)0, acc, false, false);
    }
  }

  // D column 0: lane 0 holds M=0..7 (vgpr j), lane 16 holds M=8..15
  if (lane == 0u) {
#pragma unroll
    for (int j = 0; j < 8; ++j) smem[RS_OFF + wv * 16 + j] = acc[j];
  }
  if (lane == 16u) {
#pragma unroll
    for (int j = 0; j < 8; ++j) smem[RS_OFF + wv * 16 + 8 + j] = acc[j];
  }
  __syncthreads();

  if (tid < 16u) {
    float s = smem[RS_OFF + tid] + smem[RS_OFF + 16 + tid] +
              smem[RS_OFF + 32 + tid] + smem[RS_OFF + 48 + tid] +
              bias[orow0 + tid];
    out[(size_t)b * OUT_ + orow0 + tid] = s;
  }
}

extern "C" void kernel_launch(void* const* d_in, const int* in_sizes, int n_in,
                              void* d_out, int out_size, void* d_ws, size_t ws_size,
                              hipStream_t stream) {
  const float* x    = (const float*)d_in[0];
  const float* u    = (const float*)d_in[1];
  const float* w    = (const float*)d_in[2];
  const float* la   = (const float*)d_in[3];
  const float* bias = (const float*)d_in[4];
  float* out = (float*)d_out;

  dim3 grid(B_ * (OUT_ / 16));  // 4096 blocks: (b, 16-row o-tile)
  dim3 block(128);              // 4 waves, one i-quarter each
  hipLaunchKernelGGL(l0_masked_linear_kernel, grid, block, 0, stream,
                     x, u, w, la, bias, out);
}